// AhaDiffuser_79474074845631
// MI455X (gfx1250) — compile-verified
//
#include <hip/hip_runtime.h>
#include <hip/hip_bf16.h>
#include <math.h>

// MI455X / gfx1250 implementation.
// Only the last timestep matters (output = b[:, -1, :]; all ops pointwise in t),
// so the problem collapses to dual-row (2-token) GEMVs over the weight tensors:
// facet_w 256MB + state_w 32MB + compress_w 4MB of fp32 streaming -> memory bound
// (~13us at 23.3 TB/s). We run the GEMVs on V_WMMA_F32_16X16X4_F32 with the two
// activation rows padded into a 16x4 A tile, splitting D into 8 chunks for
// latency-hiding parallelism, with deterministic partial-sum reduction.

typedef __attribute__((ext_vector_type(2))) float v2f;
typedef __attribute__((ext_vector_type(8))) float v8f;

#define DCHUNK 128          // D (=1024) split into 8 chunks
#define WAVES_PER_BLOCK 8   // 256 threads, each wave owns a 16-col N tile

// partial[s][k][r][n] = sum_{d in chunk s} A[r][d] * W[k][d][n],  r in {0,1}
__global__ __launch_bounds__(256) void dualrow_wmma_gemm(
    const float* __restrict__ row0,     // A row 0 (length Dtot)
    const float* __restrict__ row1,     // A row 1
    const float* __restrict__ W,        // [Kdim][Dtot][N], n fastest
    float* __restrict__ partials,       // [S][Kdim][2][N]
    int N, long long strideK, int Kdim)
{
    __shared__ float sha[2 * DCHUNK];
    const int tid = threadIdx.x;
    const int k  = blockIdx.y;
    const int s  = blockIdx.z;
    const int d0 = s * DCHUNK;

    // preload the two activation-row chunks into LDS
    {
        int r = tid >> 7;            // 0..1
        int dd = tid & 127;          // 0..127
        const float* rp = r ? row1 : row0;
        sha[r * DCHUNK + dd] = rp[d0 + dd];
    }
    __syncthreads();

    const int wave = tid >> 5;
    const int lane = tid & 31;
    const int m  = lane & 15;        // row (A/D) or column-within-tile (B)
    const int hsel = lane >> 4;      // half-wave select (K offset 0 vs 2)
    const int n0 = (blockIdx.x * WAVES_PER_BLOCK + wave) * 16;

    const float* Wk = W + (long long)k * strideK;
    // lane's B column pointer: rows d0+2*hsel, +1; column n0+m
    const float* wp = Wk + (long long)(d0 + 2 * hsel) * N + (n0 + m);

    // A rows >= 2 are zero padding
    const int   arow  = (m < 2) ? m : 0;
    const float amask = (m < 2) ? 1.0f : 0.0f;

    v8f c = {};
#pragma unroll 4
    for (int dd = 0; dd < DCHUNK; dd += 4) {
        // A fragment (16x4): lanes 0-15 hold K=dd,dd+1; lanes 16-31 K=dd+2,dd+3
        float a0 = sha[arow * DCHUNK + dd + 2 * hsel]     * amask;
        float a1 = sha[arow * DCHUNK + dd + 2 * hsel + 1] * amask;
        v2f a = { a0, a1 };
        // B fragment (4x16): VGPR0 rows dd(+2), VGPR1 rows dd+1(+3), col n0+m
        v2f bfrag = { wp[0], wp[N] };
        c = __builtin_amdgcn_wmma_f32_16x16x4_f32(
                /*neg_a=*/false, a, /*neg_b=*/false, bfrag,
                /*c_mod=*/(short)0, c, /*reuse_a=*/false, /*reuse_b=*/false);
        wp += (long long)4 * N;
    }

    // D rows 0,1 live in c[0], c[1] of lanes 0-15
    if (lane < 16) {
        long long base = (((long long)s * Kdim + k) * 2) * (long long)N;
        partials[base + (n0 + lane)]     = c[0];
        partials[base + N + (n0 + lane)] = c[1];
    }
}

// Reduce facet partials over D-chunks, add bias, compute per-(b,k)
// logp = logit[target] - logsumexp_V(logits).  One block per (b,k).
__global__ __launch_bounds__(1024) void facet_lse(
    const float* __restrict__ fpart,    // [8][8][2][8192]
    const float* __restrict__ facet_b,  // [8][8192]
    const int*   __restrict__ targets,  // [B][T]
    float* __restrict__ logp,           // [2][8]
    int T)
{
    const int V = 8192, K = 8, S = 8;
    __shared__ float red[1024];
    __shared__ float sh_t;
    const int b = blockIdx.x >> 3;
    const int k = blockIdx.x & 7;
    const int tid = threadIdx.x;
    const int tgt = targets[b * T + (T - 1)];

    float vals[8];
#pragma unroll
    for (int i = 0; i < 8; ++i) {
        int v = tid + i * 1024;
        float acc = facet_b[k * V + v];
#pragma unroll
        for (int s = 0; s < S; ++s)
            acc += fpart[(((long long)s * K + k) * 2 + b) * V + v];
        vals[i] = acc;
        if (v == tgt) sh_t = acc;
    }
    float mloc = vals[0];
#pragma unroll
    for (int i = 1; i < 8; ++i) mloc = fmaxf(mloc, vals[i]);
    red[tid] = mloc; __syncthreads();
    for (int off = 512; off > 0; off >>= 1) {
        if (tid < off) red[tid] = fmaxf(red[tid], red[tid + off]);
        __syncthreads();
    }
    float mx = red[0]; __syncthreads();

    float sloc = 0.f;
#pragma unroll
    for (int i = 0; i < 8; ++i) sloc += expf(vals[i] - mx);
    red[tid] = sloc; __syncthreads();
    for (int off = 512; off > 0; off >>= 1) {
        if (tid < off) red[tid] += red[tid + off];
        __syncthreads();
    }
    if (tid == 0) logp[b * 8 + k] = sh_t - (mx + logf(red[0]));
}

__global__ __launch_bounds__(256) void states_reduce(
    const float* __restrict__ spart,    // [8][8][2][1024]
    const float* __restrict__ state_b,  // [8][1024]
    float* __restrict__ states)         // [2][8][1024]
{
    const int N = 1024, K = 8, S = 8;
    int k = blockIdx.x, b = blockIdx.y, tid = threadIdx.x;
#pragma unroll
    for (int j = 0; j < 4; ++j) {
        int n = tid + j * 256;
        float acc = state_b[k * N + n];
#pragma unroll
        for (int s = 0; s < S; ++s)
            acc += spart[(((long long)s * K + k) * 2 + b) * N + n];
        states[((long long)b * K + k) * N + n] = acc;
    }
}

__global__ __launch_bounds__(256) void compress_reduce(
    const float* __restrict__ cpart,      // [8][1][2][1024]
    const float* __restrict__ compress_b, // [1024]
    float* __restrict__ out)              // [2][1024]
{
    const int N = 1024, S = 8;
    int b = blockIdx.x, tid = threadIdx.x;
#pragma unroll
    for (int j = 0; j < 4; ++j) {
        int n = tid + j * 256;
        float acc = compress_b[n];
#pragma unroll
        for (int s = 0; s < S; ++s)
            acc += cpart[((long long)s * 2 + b) * N + n];
        out[b * N + n] = acc;
    }
}

// Per-token: gate softmaxes, top-2, aha/boost logic, state mix, layernorm.
__global__ __launch_bounds__(256) void final_mix(
    const float* __restrict__ h,
    const float* __restrict__ em_w,  const float* __restrict__ em_b,
    const float* __restrict__ mfs_w, const float* __restrict__ mfs_b,
    const float* __restrict__ logp,     // [2][8]
    const float* __restrict__ states,   // [2][8][1024]
    const float* __restrict__ ln_scale, const float* __restrict__ ln_bias,
    float* __restrict__ xln)            // [2][1024]
{
    const int T = 1024, D = 1024, K = 8;
    __shared__ float red[256];
    __shared__ float shE[8], shM[8], shboost[8], shstat[2];
    const int b = blockIdx.x, tid = threadIdx.x;
    const float* hb = h + ((long long)b * T + (T - 1)) * D;

    float accE[K], accM[K];
#pragma unroll
    for (int k = 0; k < K; ++k) { accE[k] = 0.f; accM[k] = 0.f; }
    for (int d = tid; d < D; d += 256) {
        float hv = hb[d];
#pragma unroll
        for (int k = 0; k < K; ++k) {
            accE[k] += hv * em_w[d * K + k];
            accM[k] += hv * mfs_w[d * K + k];
        }
    }
    for (int k = 0; k < K; ++k) {
        red[tid] = accE[k]; __syncthreads();
        for (int off = 128; off > 0; off >>= 1) {
            if (tid < off) red[tid] += red[tid + off];
            __syncthreads();
        }
        if (tid == 0) shE[k] = red[0] + em_b[k];
        __syncthreads();
        red[tid] = accM[k]; __syncthreads();
        for (int off = 128; off > 0; off >>= 1) {
            if (tid < off) red[tid] += red[tid + off];
            __syncthreads();
        }
        if (tid == 0) shM[k] = red[0] + mfs_b[k];
        __syncthreads();
    }
    if (tid == 0) {
        float G[8], g[8], lp[8];
        float mg = shE[0]; for (int k = 1; k < K; ++k) mg = fmaxf(mg, shE[k]);
        float sg = 0.f; for (int k = 0; k < K; ++k) { G[k] = expf(shE[k] - mg); sg += G[k]; }
        for (int k = 0; k < K; ++k) G[k] /= sg;
        float mm = shM[0]; for (int k = 1; k < K; ++k) mm = fmaxf(mm, shM[k]);
        float sm = 0.f; for (int k = 0; k < K; ++k) { g[k] = expf(shM[k] - mm); sm += g[k]; }
        for (int k = 0; k < K; ++k) g[k] /= sm;
        for (int k = 0; k < K; ++k) lp[k] = logp[b * K + k];
        // top-2 of G (first occurrence on ties, like lax.top_k)
        int i1 = 0; for (int k = 1; k < K; ++k) if (G[k] > G[i1]) i1 = k;
        int i2 = (i1 == 0) ? 1 : 0;
        for (int k = 0; k < K; ++k) if (k != i1 && G[k] > G[i2]) i2 = k;
        // log_mix = logsumexp_k(log(max(g,1e-9)) + logp_k)
        float z[8], zm = -3.4e38f;
        for (int k = 0; k < K; ++k) { z[k] = logf(fmaxf(g[k], 1e-9f)) + lp[k]; zm = fmaxf(zm, z[k]); }
        float zs = 0.f; for (int k = 0; k < K; ++k) zs += expf(z[k] - zm);
        float log_mix = zm + logf(zs);
        bool any = false; float boosted[8];
        for (int k = 0; k < K; ++k) {
            float s = lp[k] - log_mix;
            bool sel = (k == i1) || (k == i2);
            bool aha = (s > 0.7f) && !sel;
            any = any || aha;
            boosted[k] = G[k] * (aha ? 2.0f : 1.0f);
        }
        if (any) boosted[i1] += 0.5f;    // MAX_PAIRS=1: +0.5 at top-1
        float bs = 0.f; for (int k = 0; k < K; ++k) bs += boosted[k];
        bs = fmaxf(bs, 1e-9f);
        for (int k = 0; k < K; ++k) shboost[k] = boosted[k] / bs;
    }
    __syncthreads();

    float x[4];
#pragma unroll
    for (int j = 0; j < 4; ++j) {
        int e = tid + j * 256;
        float acc = 0.f;
#pragma unroll
        for (int k = 0; k < K; ++k)
            acc += shboost[k] * states[((long long)b * K + k) * D + e];
        x[j] = acc;
    }
    float ls = x[0] + x[1] + x[2] + x[3];
    red[tid] = ls; __syncthreads();
    for (int off = 128; off > 0; off >>= 1) {
        if (tid < off) red[tid] += red[tid + off];
        __syncthreads();
    }
    if (tid == 0) shstat[0] = red[0] / (float)D;
    __syncthreads();
    float mu = shstat[0];
    float lv = 0.f;
#pragma unroll
    for (int j = 0; j < 4; ++j) { float d = x[j] - mu; lv += d * d; }
    red[tid] = lv; __syncthreads();
    for (int off = 128; off > 0; off >>= 1) {
        if (tid < off) red[tid] += red[tid + off];
        __syncthreads();
    }
    if (tid == 0) shstat[1] = red[0] / (float)D;
    __syncthreads();
    float inv = rsqrtf(shstat[1] + 1e-5f);
#pragma unroll
    for (int j = 0; j < 4; ++j) {
        int e = tid + j * 256;
        xln[(long long)b * D + e] = (x[j] - mu) * inv * ln_scale[e] + ln_bias[e];
    }
}

extern "C" void kernel_launch(void* const* d_in, const int* in_sizes, int n_in,
                              void* d_out, int out_size, void* d_ws, size_t ws_size,
                              hipStream_t stream) {
    (void)in_sizes; (void)n_in; (void)out_size; (void)ws_size;
    const float* h          = (const float*)d_in[0];
    const int*   targets    = (const int*)  d_in[1];
    const float* em_w       = (const float*)d_in[2];
    const float* em_b       = (const float*)d_in[3];
    const float* state_w    = (const float*)d_in[4];
    const float* state_b    = (const float*)d_in[5];
    const float* mfs_w      = (const float*)d_in[6];
    const float* mfs_b      = (const float*)d_in[7];
    const float* facet_w    = (const float*)d_in[8];
    const float* facet_b    = (const float*)d_in[9];
    const float* ln_scale   = (const float*)d_in[10];
    const float* ln_bias    = (const float*)d_in[11];
    const float* compress_w = (const float*)d_in[12];
    const float* compress_b = (const float*)d_in[13];
    float* out = (float*)d_out;

    const int B = 2, T = 1024, D = 1024, K = 8, V = 8192, S = 8;

    float* ws     = (float*)d_ws;
    float* fpart  = ws;                                  // S*K*2*V   = 1,048,576
    float* spart  = fpart + (long long)S * K * 2 * V;    // S*K*2*D   =   131,072
    float* cpart  = spart + (long long)S * K * 2 * D;    // S*1*2*D   =    16,384
    float* states = cpart + (long long)S * 2 * D;        // 2*K*D     =    16,384
    float* logp   = states + (long long)2 * K * D;       // 2*K       =        16
    float* xln    = logp + 2 * K;                        // 2*D       =     2,048

    const float* h0 = h + ((long long)0 * T + (T - 1)) * D;
    const float* h1 = h + ((long long)1 * T + (T - 1)) * D;

    // 1) logits partials: 256MB facet_w streamed once, shared across both tokens
    dualrow_wmma_gemm<<<dim3(V / 128, K, S), 256, 0, stream>>>(
        h0, h1, facet_w, fpart, V, (long long)D * V, K);
    // 2) states partials: 32MB state_w
    dualrow_wmma_gemm<<<dim3(D / 128, K, S), 256, 0, stream>>>(
        h0, h1, state_w, spart, D, (long long)D * D, K);
    // 3) logsumexp over V + target gather -> logp[2][8]
    facet_lse<<<16, 1024, 0, stream>>>(fpart, facet_b, targets, logp, T);
    // 4) states reduce (+bias)
    states_reduce<<<dim3(K, B), 256, 0, stream>>>(spart, state_b, states);
    // 5) gates / aha / boost / mix / layernorm -> xln
    final_mix<<<B, 256, 0, stream>>>(h, em_w, em_b, mfs_w, mfs_b,
                                     logp, states, ln_scale, ln_bias, xln);
    // 6) compress GEMM on WMMA, 7) reduce (+bias) -> out
    dualrow_wmma_gemm<<<dim3(D / 128, 1, S), 256, 0, stream>>>(
        xln, xln + D, compress_w, cpart, D, 0LL, 1);
    compress_reduce<<<B, 256, 0, stream>>>(cpart, compress_b, out);
}